// Sampler_73529840107542
// MI455X (gfx1250) — compile-verified
//
#include <hip/hip_runtime.h>
#include <math.h>

// ---------------------------------------------------------------------------
// CDNA5 (gfx1250) sampler: temperature -> top-k -> top-p -> min-p -> sample.
// Selection via 4096-bin order-key histogram (2 streaming passes) instead of
// a full per-row sort. Data movement uses the Tensor Data Mover (async DMA
// global->LDS, TENSORcnt) with double-buffered 32KB tiles.
// NOTE: scalar (b32) tile consumption on purpose -- the float4/ds_load_b128
// variant reproducibly crashes this toolchain's LTO ISel (constrainRegClass
// segfault in LLVM 23.1-rc2); this exact form is verified to compile.
// ---------------------------------------------------------------------------

#define NT   256          // threads per block (8 wave32)
#define TILE 8192         // floats per TDM tile (32KB)
#define NB   4096         // histogram bins (top 12 bits of order key)
#define CH   (NB / NT)    // bins per thread in scan phase (16)

#if __has_builtin(__builtin_amdgcn_tensor_load_to_lds)
#define USE_TDM 1
#else
#define USE_TDM 0
#endif

typedef unsigned int u32x4 __attribute__((ext_vector_type(4)));
typedef int          i32x8 __attribute__((ext_vector_type(8)));
typedef int          i32x4 __attribute__((ext_vector_type(4)));

// Low 32 bits of the LDS-aperture flat address == LDS byte offset (ISA 10.2).
__device__ __forceinline__ unsigned lds_off(const void* p) {
  return (unsigned)(unsigned long long)p;
}

// Order-preserving key: ascending key <-> ascending float.
__device__ __forceinline__ unsigned floatKey(float x) {
  unsigned u = __float_as_uint(x);
  return (u & 0x80000000u) ? ~u : (u | 0x80000000u);
}
__device__ __forceinline__ float keyToFloat(unsigned key) {
  unsigned u = (key & 0x80000000u) ? (key & 0x7fffffffu) : ~key;
  return __uint_as_float(u);
}

// Counter-based RNG -> Gumbel noise; argmax(logit+gumbel) == categorical.
__device__ __forceinline__ float gumbel_noise(unsigned row, unsigned idx) {
  unsigned h = idx * 0x9E3779B1u + row * 0x85EBCA77u + 42u * 0x27D4EB2Fu;
  h ^= h >> 16; h *= 0x7FEB352Du;
  h ^= h >> 15; h *= 0x846CA68Bu;
  h ^= h >> 16;
  float u = (float)(h >> 8) * 5.9604645e-08f + 1.4901161e-08f;  // (0,1)
  return -__logf(-__logf(u));
}

// TDM: 1-D tile load, D# built per CDNA5 ISA.
// tensor_dim0 = valid element count so the tail tile zero-fills (OOB reads
// return zero); processing is bounded by `valid` anyway.
__device__ __forceinline__ void tdm_load_row_tile(unsigned lds_byte, const float* g,
                                                  unsigned valid, unsigned tile) {
#if USE_TDM
  unsigned long long ga = (unsigned long long)(const void*)g;
  u32x4 g0;
  g0[0] = 1u;                                              // count=1, user desc
  g0[1] = lds_byte;                                        // lds_addr
  g0[2] = (unsigned)(ga & 0xffffffffull);                  // global_addr[31:0]
  g0[3] = (unsigned)((ga >> 32) & 0x01ffffffull) | 0x80000000u; // [56:32] | type=2
  i32x8 g1;
  g1[0] = 0x00020000;                                      // data_size = 4B
  g1[1] = (int)((valid & 0xffffu) << 16);                  // tensor_dim0[15:0]
  g1[2] = (int)((valid >> 16) | (1u << 16));               // dim0[31:16] | tensor_dim1=1
  g1[3] = (int)(tile << 16);                               // tile_dim0
  g1[4] = 1;                                               // tile_dim1 = 1
  g1[5] = (int)tile;                                       // tensor_dim0_stride
  g1[6] = 0;
  g1[7] = 0;
  i32x4 z4 = {0, 0, 0, 0};
#if defined(__clang_major__) && (__clang_major__ >= 23)
  i32x8 z8 = {0, 0, 0, 0, 0, 0, 0, 0};
  __builtin_amdgcn_tensor_load_to_lds(g0, g1, z4, z4, z8, 0);
#else
  __builtin_amdgcn_tensor_load_to_lds(g0, g1, z4, z4, 0);
#endif
#else
  (void)lds_byte; (void)g; (void)valid; (void)tile;
#endif
}

__device__ __forceinline__ void wait_tensorcnt_le1() {
#if __has_builtin(__builtin_amdgcn_s_wait_tensorcnt)
  __builtin_amdgcn_s_wait_tensorcnt(1);
#endif
}
__device__ __forceinline__ void wait_tensorcnt_0() {
#if __has_builtin(__builtin_amdgcn_s_wait_tensorcnt)
  __builtin_amdgcn_s_wait_tensorcnt(0);
#endif
}

// Stream one row through double-buffered LDS tiles; body(idx, value).
template <class F>
__device__ __forceinline__ void stream_row(const float* rowPtr, int V, int tid,
                                           float (*sTile)[TILE], F&& body) {
  const int nTiles = (V + TILE - 1) / TILE;
#if USE_TDM
  if (tid < 32)  // wave 0 owns the DMA queue (TENSORcnt is per-wave)
    tdm_load_row_tile(lds_off(&sTile[0][0]), rowPtr,
                      (unsigned)min(V, TILE), (unsigned)TILE);
#endif
  for (int t = 0; t < nTiles; ++t) {
    const int base  = t * TILE;
    const int valid = min(TILE, V - base);
#if USE_TDM
    if (tid < 32) {
      const int nx = t + 1;
      if (nx < nTiles) {
        tdm_load_row_tile(lds_off(&sTile[nx & 1][0]), rowPtr + nx * TILE,
                          (unsigned)min(V - nx * TILE, TILE), (unsigned)TILE);
        wait_tensorcnt_le1();   // 2 in flight -> wait until tile t landed
      } else {
        wait_tensorcnt_0();
      }
    }
    __syncthreads();            // TDM data visible to all 8 waves
    const float* cur = sTile[t & 1];
#else
    for (int i = tid; i < valid; i += NT) sTile[0][i] = rowPtr[base + i];
    __syncthreads();
    const float* cur = sTile[0];
#endif
    for (int i = tid; i < valid; i += NT) body(base + i, cur[i]);
    __syncthreads();            // tile consumed before buffer is reused
  }
}

__global__ void __launch_bounds__(NT)
sampler_kernel(const float* __restrict__ logits, const float* __restrict__ temps,
               const float* __restrict__ minps, const float* __restrict__ topps,
               const int* __restrict__ topks, int* __restrict__ out, int V) {
  __shared__ float              sTile[2][TILE];   // 64KB DMA staging
  __shared__ unsigned           sCount[NB];       // 16KB per-bin counts
  __shared__ unsigned long long sMass[NB];        // 32KB per-bin fixed-point mass
  __shared__ float              sAbs[NB];         // 16KB per-bin absolute mass
  __shared__ float              sChunkM[NT];
  __shared__ unsigned           sChunkC[NT];
  __shared__ float              sRedF[NT];
  __shared__ int                sRedI[NT];
  __shared__ float              s_m, s_Zp, s_minpT;
  __shared__ unsigned           s_bk, s_bp, s_keyThresh;

  const int tid = threadIdx.x;
  const int row = blockIdx.x;
  const float* rowPtr = logits + (long long)row * V;
  const float invT = 1.0f / temps[row];
  const float minp = minps[row];
  const float topp = topps[row];
  const unsigned k = (unsigned)max(topks[row], 1);

  for (int i = tid; i < NB; i += NT) { sCount[i] = 0u; sMass[i] = 0ull; }
  if (tid == 0) { s_bk = 0xFFFFFFFFu; s_bp = 0xFFFFFFFFu; }
  __syncthreads();

  // ---- Pass 1: key histogram (count + exp-mass) and exact row max ---------
  // Mass stored relative to each bin's scaled upper edge -> exp arg <= ~0,
  // no overflow, no need for the global max during accumulation. Fixed-point
  // u64 LDS atomics keep the sums order-independent (deterministic replays).
  float rawMax = -INFINITY;
  stream_row(rowPtr, V, tid, sTile, [&](int idx, float x) {
    (void)idx;
    rawMax = fmaxf(rawMax, x);
    const unsigned key = floatKey(x);
    const unsigned b = key >> 20;
    atomicAdd(&sCount[b], 1u);
    const float ref = keyToFloat(((b + 1u) << 20) - 1u) * invT; // bin top, scaled
    const float e = __expf(x * invT - ref);                     // in (0, 1]
    atomicAdd(&sMass[b], (unsigned long long)(e * 4294967296.0f));
  });

  sRedF[tid] = rawMax;
  __syncthreads();
  for (int s = NT / 2; s > 0; s >>= 1) {
    if (tid < s) sRedF[tid] = fmaxf(sRedF[tid], sRedF[tid + s]);
    __syncthreads();
  }
  if (tid == 0) s_m = sRedF[0] * invT;   // max of scaled logits (invT > 0)
  __syncthreads();
  const float m = s_m;

  // ---- Bin scan: per-chunk sums in descending bin order -------------------
  float cm = 0.f; unsigned cc = 0u;
  for (int j = 0; j < CH; ++j) {
    const int d = tid * CH + j;
    const unsigned b = (unsigned)(NB - 1 - d);
    const unsigned cnt = sCount[b];
    float am = 0.f;
    if (cnt) {
      const float ref = keyToFloat(((b + 1u) << 20) - 1u) * invT;
      am = (float)sMass[b] * 2.3283064365386963e-10f * __expf(ref - m);
    }
    sAbs[b] = am;
    cm += am; cc += cnt;
  }
  sChunkM[tid] = cm; sChunkC[tid] = cc;
  __syncthreads();
  if (tid == 0) {     // 256-entry exclusive prefix (from the top bin down)
    float pm = 0.f; unsigned pc = 0u;
    for (int t = 0; t < NT; ++t) {
      const float a = sChunkM[t]; const unsigned c = sChunkC[t];
      sChunkM[t] = pm; sChunkC[t] = pc;
      pm += a; pc += c;
    }
  }
  __syncthreads();

  // ---- top-k cut: lowest bin b with countAbove(b) < k ---------------------
  {
    unsigned above = sChunkC[tid], minKept = 0xFFFFFFFFu;
    for (int j = 0; j < CH; ++j) {
      const unsigned b = (unsigned)(NB - 1 - (tid * CH + j));
      if (above < k) minKept = b;
      above += sCount[b];
    }
    if (minKept != 0xFFFFFFFFu) atomicMin(&s_bk, minKept);
  }
  __syncthreads();

  // ---- Z' = softmax mass of bins surviving the top-k cut ------------------
  {
    const unsigned bk = s_bk;
    const int d = NB - 1 - (int)bk;
    if (tid == d / CH) {
      float above = sChunkM[tid];
      const int jl = d & (CH - 1);
      for (int j = 0; j < jl; ++j) above += sAbs[NB - 1 - (tid * CH + j)];
      s_Zp = above + sAbs[bk];
    }
  }
  __syncthreads();

  // ---- top-p cut: keep bin b while massAbove(b) < p * Z' ------------------
  // (token kept iff suffix-mass of strictly-larger tokens < p*Z', which is
  //  exactly the reference's "cumsum > 1-p"; max token always survives)
  {
    const float target = topp * s_Zp;
    float above = sChunkM[tid];
    unsigned minKept = 0xFFFFFFFFu;
    for (int j = 0; j < CH; ++j) {
      const unsigned b = (unsigned)(NB - 1 - (tid * CH + j));
      if (above < target) minKept = b;
      above += sAbs[b];
    }
    if (minKept != 0xFFFFFFFFu) atomicMin(&s_bp, minKept);
  }
  __syncthreads();
  if (tid == 0) {
    const unsigned bt = (s_bk > s_bp) ? s_bk : s_bp;
    s_keyThresh = bt << 20;                                  // bin lower edge
    s_minpT = (minp > 0.f) ? (m + __logf(minp)) : -INFINITY; // min-p in logit space
  }
  __syncthreads();
  const unsigned keyThresh = s_keyThresh;
  const float minpT = s_minpT;

  // ---- Pass 2: Gumbel-max categorical sample over surviving tokens --------
  float best = -INFINITY; int bestIdx = 0;
  stream_row(rowPtr, V, tid, sTile, [&](int idx, float x) {
    const unsigned key = floatKey(x);
    const float sc = x * invT;
    if (key >= keyThresh && sc >= minpT) {
      const float g = sc + gumbel_noise((unsigned)row, (unsigned)idx);
      if (g > best || (g == best && idx < bestIdx)) { best = g; bestIdx = idx; }
    }
  });

  sRedF[tid] = best; sRedI[tid] = bestIdx;
  __syncthreads();
  for (int s = NT / 2; s > 0; s >>= 1) {
    if (tid < s) {
      const float o = sRedF[tid + s]; const int oi = sRedI[tid + s];
      if (o > sRedF[tid] || (o == sRedF[tid] && oi < sRedI[tid])) {
        sRedF[tid] = o; sRedI[tid] = oi;
      }
    }
    __syncthreads();
  }
  if (tid == 0) out[row] = sRedI[0];
}

extern "C" void kernel_launch(void* const* d_in, const int* in_sizes, int n_in,
                              void* d_out, int out_size, void* d_ws, size_t ws_size,
                              hipStream_t stream) {
  (void)n_in; (void)d_ws; (void)ws_size; (void)out_size;
  const float* logits = (const float*)d_in[0];
  const float* temps  = (const float*)d_in[1];
  const float* minps  = (const float*)d_in[2];
  const float* topps  = (const float*)d_in[3];
  const int*   topks  = (const int*)d_in[4];
  const int B = in_sizes[1];
  const int V = in_sizes[0] / B;
  sampler_kernel<<<B, NT, 0, stream>>>(logits, temps, minps, topps, topks,
                                       (int*)d_out, V);
}